// Word2VecEmbedding_20100446945520
// MI455X (gfx1250) — compile-verified
//
#include <hip/hip_runtime.h>

// Word2Vec embedding with UNK -> context-average replacement.
// Memory-bound gather (no matrix FLOPs -> WMMA inapplicable by roofline).
// Optimized for MI455X: 16B-aligned b128 traffic, wave32 block sizing,
// uniform scalar id loads, gfx1250 global_prefetch_b8 on the gather rows.

namespace {
constexpr int kDim   = 300;        // embedding dim
constexpr int kVec   = kDim / 4;   // 75 float4 per row (1200 B, 16B aligned)
constexpr int kSeq   = 4096;       // sequence length (power of two)
constexpr int kPad   = 0;
constexpr int kUnk   = 1;
constexpr int kCtx   = 2;          // context window radius
constexpr int kWin   = 2 * kCtx + 1;
constexpr int kBlock = 96;         // 3 x wave32; lanes 0..74 carry the payload
}

__global__ __launch_bounds__(kBlock)
void w2v_embed_ctx_avg_kernel(const int*   __restrict__ ids,
                              const float* __restrict__ table,
                              float*       __restrict__ out) {
  const int pos = blockIdx.x;            // flat token index in [0, B*S)
  const int t   = threadIdx.x;
  const int s   = pos & (kSeq - 1);      // position within sequence

  const int id = ids[pos];               // block-uniform -> scalar load
  float4* __restrict__ orow = reinterpret_cast<float4*>(out) + (size_t)pos * kVec;

  if (id != kUnk) {
    // ~95% path: straight row copy (PAD id copies the zero row, as reference).
    if (t < kVec) {
      const float4* __restrict__ row =
          reinterpret_cast<const float4*>(table) + (size_t)id * kVec;
      orow[t] = row[t];                  // global_load_b128 + global_store_b128
    }
    return;
  }

  // UNK: masked average over the +/-2 window. Center is UNK (invalid) and
  // out-of-sequence neighbors are zero-padded (invalid) -> skipped.
  int   nid[kWin];
  bool  val[kWin];
  float cnt = 0.0f;
#pragma unroll
  for (int k = 0; k < kWin; ++k) {
    const int  j  = s + k - kCtx;
    const bool in = (unsigned)j < (unsigned)kSeq;
    const int  v  = in ? ids[pos + k - kCtx] : kPad;   // uniform scalar loads
    nid[k] = v;
    val[k] = in && (v != kPad) && (v != kUnk);
    cnt   += val[k] ? 1.0f : 0.0f;
  }

  if (t < kVec) {
    const float4* __restrict__ tab4 = reinterpret_cast<const float4*>(table);

    // Warm the gathered rows (L2-resident table): gfx1250 global_prefetch_b8.
#pragma unroll
    for (int k = 0; k < kWin; ++k)
      if (val[k]) __builtin_prefetch(tab4 + (size_t)nid[k] * kVec + t, 0, 3);

    float ax = 0.0f, ay = 0.0f, az = 0.0f, aw = 0.0f;
#pragma unroll
    for (int k = 0; k < kWin; ++k) {
      if (val[k]) {                                   // scalar (uniform) branch
        const float4 e = tab4[(size_t)nid[k] * kVec + t];
        ax += e.x; ay += e.y; az += e.z; aw += e.w;
      }
    }

    const float denom = cnt + 1e-8f;                  // matches reference eps
    float4 o;
    o.x = ax / denom; o.y = ay / denom; o.z = az / denom; o.w = aw / denom;
    orow[t] = o;
  }
}

extern "C" void kernel_launch(void* const* d_in, const int* in_sizes, int n_in,
                              void* d_out, int out_size, void* d_ws, size_t ws_size,
                              hipStream_t stream) {
  (void)n_in; (void)out_size; (void)d_ws; (void)ws_size;
  const int*   ids   = reinterpret_cast<const int*>(d_in[0]);    // input_ids [B*S]
  const float* table = reinterpret_cast<const float*>(d_in[1]);  // [VOCAB, 300]
  float*       out   = reinterpret_cast<float*>(d_out);          // [B*S, 300]

  const int n_pos = in_sizes[0];   // B*S = 131072; one block per token
  w2v_embed_ctx_avg_kernel<<<n_pos, kBlock, 0, stream>>>(ids, table, out);
}